// Inverse3DMatrix_13864154432414
// MI455X (gfx1250) — compile-verified
//
#include <hip/hip_runtime.h>
#include <hip/hip_bf16.h>

// Batched inverse of [R|t] affine matrices (bottom row [0,0,0,1] implied):
//   inv = [ R^-1 | -R^-1 * t ]
// Pure streaming kernel: 48B in / 48B out per matrix, ~55 FLOPs.
// Memory-bound: 96 MB total @ 23.3 TB/s => ~4.1 us floor on MI455X.

typedef float v4f __attribute__((ext_vector_type(4)));

__global__ __launch_bounds__(256) void Inverse3DMatrix_kernel(
    const v4f* __restrict__ in, v4f* __restrict__ out, int n)
{
    int idx = blockIdx.x * blockDim.x + threadIdx.x;
    if (idx >= n) return;

    const size_t base = (size_t)idx * 3;

    // Streaming loads: each byte read exactly once -> non-temporal hint (TH=NT).
    v4f r0 = __builtin_nontemporal_load(&in[base + 0]); // a b c tx
    v4f r1 = __builtin_nontemporal_load(&in[base + 1]); // d e f ty
    v4f r2 = __builtin_nontemporal_load(&in[base + 2]); // g h i tz

    const float a = r0.x, b = r0.y, c = r0.z, tx = r0.w;
    const float d = r1.x, e = r1.y, f = r1.z, ty = r1.w;
    const float g = r2.x, h = r2.y, i = r2.z, tz = r2.w;

    // Cofactors of R (first row of cofactor matrix drives the determinant).
    const float c00 = e * i - f * h;
    const float c01 = f * g - d * i;
    const float c02 = d * h - e * g;

    const float det = a * c00 + b * c01 + c * c02;
    const float s   = 1.0f / det;

    // R^-1 = adj(R)/det (adjugate = cofactor transpose), row-major rows:
    const float i00 = c00 * s;
    const float i01 = (c * h - b * i) * s;
    const float i02 = (b * f - c * e) * s;

    const float i10 = c01 * s;
    const float i11 = (a * i - c * g) * s;
    const float i12 = (c * d - a * f) * s;

    const float i20 = c02 * s;
    const float i21 = (b * g - a * h) * s;
    const float i22 = (a * e - b * d) * s;

    // Translation: t' = -(R^-1 * t)
    const float ntx = -(i00 * tx + i01 * ty + i02 * tz);
    const float nty = -(i10 * tx + i11 * ty + i12 * tz);
    const float ntz = -(i20 * tx + i21 * ty + i22 * tz);

    v4f o0 = { i00, i01, i02, ntx };
    v4f o1 = { i10, i11, i12, nty };
    v4f o2 = { i20, i21, i22, ntz };

    // Streaming stores: output never re-read on device -> non-temporal hint.
    __builtin_nontemporal_store(o0, &out[base + 0]);
    __builtin_nontemporal_store(o1, &out[base + 1]);
    __builtin_nontemporal_store(o2, &out[base + 2]);
}

extern "C" void kernel_launch(void* const* d_in, const int* in_sizes, int n_in,
                              void* d_out, int out_size, void* d_ws, size_t ws_size,
                              hipStream_t stream)
{
    (void)n_in; (void)out_size; (void)d_ws; (void)ws_size;

    const v4f* in  = (const v4f*)d_in[0];
    v4f*       out = (v4f*)d_out;

    const int n = in_sizes[0] / 12;          // number of 3x4 matrices
    const int threads = 256;                 // 8 wave32s per block
    const int blocks  = (n + threads - 1) / threads;

    Inverse3DMatrix_kernel<<<blocks, threads, 0, stream>>>(in, out, n);
}